// GATModel_36979668418676
// MI455X (gfx1250) — compile-verified
//
#include <hip/hip_runtime.h>

typedef float v2f __attribute__((ext_vector_type(2)));
typedef float v8f __attribute__((ext_vector_type(8)));

#define NEG_SLOPE 0.2f

__device__ __forceinline__ void atomicMaxF(float* addr, float v) {
    // sign-split trick: valid when initialized to -inf
    if (v >= 0.0f) atomicMax((int*)addr, __float_as_int(v));
    else           atomicMin((unsigned int*)addr, __float_as_uint(v));
}

__device__ __forceinline__ float lrelu(float v) {
    return v > 0.0f ? v : NEG_SLOPE * v;
}

// ---------------- init: zero / -inf all accumulators (every call) -------------
__global__ void gat_init(float* m1, float* s1, float* num1,
                         float* m2, float* s2, float* out, int n) {
    int i = blockIdx.x * blockDim.x + threadIdx.x;
    if (i < 4 * n) { m1[i] = -__builtin_inff(); s1[i] = 0.0f; num1[i] = 0.0f; }
    if (i < n)     { m2[i] = -__builtin_inff(); s2[i] = 0.0f; }
    if (i < 16 * n) out[i] = 0.0f;
}

// ---------------- layer-1 attention constants: S[0..3]=src, S[4..7]=dst ------
__global__ void gat_consts(const float* W1, const float* att_src1,
                           const float* att_dst1, float* S) {
    int h = threadIdx.x;
    if (h < 8) {
        const float* att = (h < 4) ? att_src1 : att_dst1;
        int hh = h & 3;
        float acc = 0.0f;
        for (int c = 0; c < 32; ++c) acc += W1[hh * 32 + c] * att[hh * 32 + c];
        S[h] = acc;
    }
}

// ---------------- layer-1 edge pass 1: segment max over dst ------------------
__global__ void gat_edge1_max(const int* ei, const float* x, const float* S,
                              float* m1, int E, int n) {
    int e = blockIdx.x * blockDim.x + threadIdx.x;
    int Et = E + n;
    if (e >= Et) return;
    int s, d;
    if (e < E) { s = ei[e]; d = ei[E + e]; } else { s = d = e - E; }
    float xs = x[s], xd = x[d];
#pragma unroll
    for (int h = 0; h < 4; ++h) {
        float v = lrelu(xs * S[h] + xd * S[4 + h]);
        atomicMaxF(&m1[d * 4 + h], v);
    }
}

// ---------------- layer-1 edge pass 2: exp-sum + weighted numerator ----------
__global__ void gat_edge1_sum(const int* ei, const float* x, const float* S,
                              const float* m1, float* s1, float* num1,
                              int E, int n) {
    int e = blockIdx.x * blockDim.x + threadIdx.x;
    int Et = E + n;
    if (e >= Et) return;
    int s, d;
    if (e < E) { s = ei[e]; d = ei[E + e]; } else { s = d = e - E; }
    float xs = x[s], xd = x[d];
#pragma unroll
    for (int h = 0; h < 4; ++h) {
        float v = lrelu(xs * S[h] + xd * S[4 + h]);
        float ev = __expf(v - m1[d * 4 + h]);
        atomicAdd(&s1[d * 4 + h], ev);
        atomicAdd(&num1[d * 4 + h], ev * xs);
    }
}

// ---------------- layer-2 GEMM: h2 = elu(h1) @ W2 via f32 WMMA ---------------
// h1[n,c] = W1[c]*t[n,c>>5] + b1[c], t = num1/s1. One wave per 16-row tile.
__global__ void gat_gemm_wmma(const float* num1, const float* s1,
                              const float* W1, const float* b1,
                              const float* W2, float* h2, int n) {
    int wave = (int)((blockIdx.x * blockDim.x + threadIdx.x) >> 5);
    int lane = threadIdx.x & 31;
    int row0 = wave * 16;
    if (row0 >= n) return;                 // wave-uniform: EXEC stays all-1s

    int m    = lane & 15;
    int koff = (lane >> 4) * 2;            // lanes 16-31 hold K+2, K+3
    int col  = lane & 15;
    int r    = row0 + m;
    int rc   = r < n ? r : n - 1;

    float t[4];
#pragma unroll
    for (int h = 0; h < 4; ++h) t[h] = num1[rc * 4 + h] / s1[rc * 4 + h];

    v8f acc = {};
    for (int k = 0; k < 128; k += 4) {
        int c0 = k + koff, c1 = c0 + 1;
        // build elu(h1) A-fragment on the fly (ISA 16x4 f32 A layout)
        float a0 = W1[c0] * t[c0 >> 5] + b1[c0];
        float a1 = W1[c1] * t[c1 >> 5] + b1[c1];
        a0 = a0 > 0.0f ? a0 : __expf(a0) - 1.0f;
        a1 = a1 > 0.0f ? a1 : __expf(a1) - 1.0f;
        v2f a; a.x = a0; a.y = a1;
        // B 4x16 fragment (row striped across lanes, halves offset by 2)
        v2f b; b.x = W2[c0 * 16 + col]; b.y = W2[c1 * 16 + col];
        acc = __builtin_amdgcn_wmma_f32_16x16x4_f32(
            false, a, false, b, (short)0, acc, false, false);
    }
    // C/D layout: VGPR v -> row v (lanes 0-15) / row v+8 (lanes 16-31)
    int rbase = row0 + ((lane >> 4) << 3);
    if (row0 + 16 <= n) {
        // full tile: branch-free coalesced stores
#pragma unroll
        for (int v = 0; v < 8; ++v) h2[(rbase + v) * 16 + col] = acc[v];
    } else {
#pragma unroll
        for (int v = 0; v < 8; ++v) {
            int rr = rbase + v;
            if (rr < n) h2[rr * 16 + col] = acc[v];
        }
    }
}

// ---------------- layer-2 per-node attention logits --------------------------
__global__ void gat_node2(const float* h2, const float* att_s, const float* att_d,
                          float* as2, float* ad2, int n) {
    int i = blockIdx.x * blockDim.x + threadIdx.x;
    if (i >= n) return;
    float a = 0.0f, b = 0.0f;
#pragma unroll
    for (int c = 0; c < 16; ++c) {
        float v = h2[i * 16 + c];
        a += v * att_s[c];
        b += v * att_d[c];
    }
    as2[i] = a; ad2[i] = b;
}

// ---------------- layer-2 edge pass 1: segment max ---------------------------
__global__ void gat_edge2_max(const int* ei, const float* as2, const float* ad2,
                              float* m2, int E, int n) {
    int e = blockIdx.x * blockDim.x + threadIdx.x;
    int Et = E + n;
    if (e >= Et) return;
    int s, d;
    if (e < E) { s = ei[e]; d = ei[E + e]; } else { s = d = e - E; }
    atomicMaxF(&m2[d], lrelu(as2[s] + ad2[d]));
}

// ---------------- layer-2 edge pass 2: exp-sum -------------------------------
__global__ void gat_edge2_sum(const int* ei, const float* as2, const float* ad2,
                              const float* m2, float* s2, int E, int n) {
    int e = blockIdx.x * blockDim.x + threadIdx.x;
    int Et = E + n;
    if (e >= Et) return;
    int s, d;
    if (e < E) { s = ei[e]; d = ei[E + e]; } else { s = d = e - E; }
    atomicAdd(&s2[d], __expf(lrelu(as2[s] + ad2[d]) - m2[d]));
}

// ---------------- layer-2 edge pass 3: scatter alpha-weighted messages -------
// one thread per (edge, channel): coalesced h2 gathers, consecutive atomics
__global__ void gat_edge2_scatter(const int* ei, const float* as2, const float* ad2,
                                  const float* m2, const float* s2, const float* h2,
                                  float* out, int E, int n) {
    long long gid = (long long)blockIdx.x * blockDim.x + threadIdx.x;
    int Et = E + n;
    if (gid >= (long long)Et * 16) return;
    int e = (int)(gid >> 4);
    int c = (int)(gid & 15);
    int s, d;
    if (e < E) { s = ei[e]; d = ei[E + e]; } else { s = d = e - E; }
    float ev    = __expf(lrelu(as2[s] + ad2[d]) - m2[d]);
    float alpha = ev / s2[d];
    atomicAdd(&out[d * 16 + c], alpha * h2[s * 16 + c]);
}

// ---------------- finish: add bias -------------------------------------------
__global__ void gat_finish(float* out, const float* b2, int n) {
    int i = blockIdx.x * blockDim.x + threadIdx.x;
    if (i < 16 * n) out[i] += b2[i & 15];
}

extern "C" void kernel_launch(void* const* d_in, const int* in_sizes, int n_in,
                              void* d_out, int out_size, void* d_ws, size_t ws_size,
                              hipStream_t stream) {
    const float* x        = (const float*)d_in[0];
    const int*   ei       = (const int*)d_in[1];
    const float* W1       = (const float*)d_in[2];
    const float* att_src1 = (const float*)d_in[3];
    const float* att_dst1 = (const float*)d_in[4];
    const float* b1       = (const float*)d_in[5];
    const float* W2       = (const float*)d_in[6];
    const float* att_src2 = (const float*)d_in[7];
    const float* att_dst2 = (const float*)d_in[8];
    const float* b2       = (const float*)d_in[9];
    float* out = (float*)d_out;

    const int n  = in_sizes[0];          // 50000 nodes
    const int E  = in_sizes[1] / 2;      // 1.6M edges (self-loops added here)
    const int Et = E + n;

    // workspace layout (floats)
    float* ws   = (float*)d_ws;
    float* S    = ws;                    // 8 (+pad to 16)
    float* m1   = ws + 16;               // 4n
    float* s1   = m1 + 4 * n;            // 4n
    float* num1 = s1 + 4 * n;            // 4n
    float* h2   = num1 + 4 * n;          // 16n
    float* as2  = h2 + 16 * n;           // n
    float* ad2  = as2 + n;               // n
    float* m2   = ad2 + n;               // n
    float* s2   = m2 + n;                // n  -> total 16 + 32n floats (~6.4MB)

    const int B = 256;
    gat_init<<<(16 * n + B - 1) / B, B, 0, stream>>>(m1, s1, num1, m2, s2, out, n);
    gat_consts<<<1, 32, 0, stream>>>(W1, att_src1, att_dst1, S);

    gat_edge1_max<<<(Et + B - 1) / B, B, 0, stream>>>(ei, x, S, m1, E, n);
    gat_edge1_sum<<<(Et + B - 1) / B, B, 0, stream>>>(ei, x, S, m1, s1, num1, E, n);

    int tiles = (n + 15) / 16;
    int gthreads = tiles * 32;
    gat_gemm_wmma<<<(gthreads + 127) / 128, 128, 0, stream>>>(num1, s1, W1, b1, W2, h2, n);

    gat_node2<<<(n + B - 1) / B, B, 0, stream>>>(h2, att_src2, att_dst2, as2, ad2, n);

    gat_edge2_max<<<(Et + B - 1) / B, B, 0, stream>>>(ei, as2, ad2, m2, E, n);
    gat_edge2_sum<<<(Et + B - 1) / B, B, 0, stream>>>(ei, as2, ad2, m2, s2, E, n);

    long long sthreads = (long long)Et * 16;
    gat_edge2_scatter<<<(unsigned)((sthreads + B - 1) / B), B, 0, stream>>>(
        ei, as2, ad2, m2, s2, h2, out, E, n);

    gat_finish<<<(16 * n + B - 1) / B, B, 0, stream>>>(out, b2, n);
}